// FocusVisionMambaBlock_37563783971172
// MI455X (gfx1250) — compile-verified
//
#include <hip/hip_runtime.h>

// ---------------- problem constants ----------------
#define BB   8
#define HH   224
#define WWID 224
#define CC   96
#define HN   112
#define WN   112
#define KDIM 384            // 4*C
#define NDIM 192            // 2*C
#define MROWS (BB*HN*WN)    // 100352
#define MTILE 16
#define NWAVES_MMA 12       // 12*16 = 192 output cols
#define THREADS 512         // 16 waves of 32
#define XM_STRIDE 388       // dwords; 388 % 64 == 4 -> conflict-free A-frag reads
#define XS_STRIDE 194       // dwords; 194 % 64 == 2 -> conflict-free C stores
#define EPSF 1e-5f

typedef __attribute__((ext_vector_type(2))) float v2f;
typedef __attribute__((ext_vector_type(4))) float v4f;
typedef __attribute__((ext_vector_type(8))) float v8f;

__device__ __forceinline__ float wave_sum32(float s) {
#pragma unroll
  for (int off = 16; off > 0; off >>= 1)
    s += __shfl_xor(s, off, 32);
  return s;
}

// LN over 192 channels held as 6 regs/lane (pairs: c = j*64 + lane*2 + t).
__device__ __forceinline__ void ln_step192(float (&v)[6], int lane,
                                           const float* __restrict__ w,
                                           const float* __restrict__ b,
                                           bool residual) {
  float s = 0.f, s2 = 0.f;
#pragma unroll
  for (int j = 0; j < 6; ++j) { s += v[j]; s2 += v[j] * v[j]; }
  s  = wave_sum32(s);
  s2 = wave_sum32(s2);
  float mu = s * (1.0f / NDIM);
  float rs = rsqrtf(s2 * (1.0f / NDIM) - mu * mu + EPSF);
#pragma unroll
  for (int j = 0; j < 3; ++j) {
    int c2 = j * 64 + lane * 2;
    v2f wv = *(const v2f*)(w + c2);
    v2f bv = *(const v2f*)(b + c2);
#pragma unroll
    for (int t = 0; t < 2; ++t) {
      float n = (v[2 * j + t] - mu) * rs * wv[t] + bv[t];
      v[2 * j + t] = residual ? (v[2 * j + t] + n) : n;
    }
  }
}

// Pre-swizzle merge_w into per-(wave,kstep,lane) float2 B-fragments so the
// GEMM inner loop does one fully-coalesced 256B global_load_b64 per wave-step.
__global__ void prep_bfrag(const float* __restrict__ mw, float2* __restrict__ bf) {
  int g    = blockIdx.x;        // 0 .. 12*96-1
  int lane = threadIdx.x;       // 0 .. 31
  int w    = g / 96;
  int step = g - w * 96;
  int col  = w * 16 + (lane & 15);
  int kk   = step * 4 + 2 * (lane >> 4);
  const float* p = mw + (size_t)col * KDIM + kk;
  float2 v; v.x = p[0]; v.y = p[1];
  bf[(size_t)g * 32 + lane] = v;
}

template <bool PREP>
__global__ __launch_bounds__(THREADS)
void fused_focus_mamba(const float* __restrict__ x,
                       const float* __restrict__ mw,
                       const float2* __restrict__ bfrag,
                       const float* __restrict__ mnw, const float* __restrict__ mnb,
                       const float* __restrict__ vssw, const float* __restrict__ vssb,
                       const float* __restrict__ asw,  const float* __restrict__ asb,
                       const float* __restrict__ fnw,  const float* __restrict__ fnb,
                       float* __restrict__ out) {
  __shared__ float xm[MTILE * XM_STRIDE];   // merged+normed A tile
  __shared__ float xs[MTILE * XS_STRIDE];   // GEMM result tile

  const int tid  = threadIdx.x;
  const int lane = tid & 31;
  const int wv   = tid >> 5;                // 0..15
  const int row0 = blockIdx.x * MTILE;

  // ---- Phase 0: gather 16 rows x 384 merged values into LDS (b128, NT) ----
  // 16*384/4 = 1536 float4s, 512 threads -> 3 each. Each 96-ch segment is
  // contiguous & 16B aligned in x; consecutive threads stream consecutive 16B.
#pragma unroll
  for (int j = 0; j < (MTILE * KDIM / 4) / THREADS; ++j) {   // 3 iters
    int idx4 = tid + j * THREADS;
    int r    = idx4 / (KDIM / 4);            // row in tile
    int i4   = idx4 - r * (KDIM / 4);        // float4 index within row
    int part = i4 / (CC / 4);                // 0..3 -> (x0,x1,x2,x3)
    int c4   = (i4 - part * (CC / 4)) * 4;   // channel
    int p    = row0 + r;
    int b    = p / (HN * WN);
    int rem  = p - b * (HN * WN);
    int hn   = rem / WN;
    int wn   = rem - hn * WN;
    int h    = 2 * hn + (part & 1);
    int w2   = 2 * wn + (part >> 1);
    const v4f* src = (const v4f*)(x + ((((size_t)b * HH + h) * WWID + w2) * CC + c4));
    v4f val = __builtin_nontemporal_load(src);
    *(v4f*)&xm[r * XM_STRIDE + part * CC + c4] = val;
  }
  __syncthreads();

  // ---- Phase 1: LayerNorm over K=384, one wave per row (float4) ----
  {
    const int r = wv;
    float s = 0.f, s2 = 0.f;
    v4f vals[3];
#pragma unroll
    for (int j = 0; j < 3; ++j) {
      vals[j] = *(const v4f*)&xm[r * XM_STRIDE + j * 128 + lane * 4];
#pragma unroll
      for (int t = 0; t < 4; ++t) { s += vals[j][t]; s2 += vals[j][t] * vals[j][t]; }
    }
    s  = wave_sum32(s);
    s2 = wave_sum32(s2);
    float mu = s * (1.0f / KDIM);
    float rs = rsqrtf(s2 * (1.0f / KDIM) - mu * mu + EPSF);
#pragma unroll
    for (int j = 0; j < 3; ++j) {
      int k4 = j * 128 + lane * 4;
      v4f wv4 = *(const v4f*)(mnw + k4);
      v4f bv4 = *(const v4f*)(mnb + k4);
      v4f o;
#pragma unroll
      for (int t = 0; t < 4; ++t) o[t] = (vals[j][t] - mu) * rs * wv4[t] + bv4[t];
      *(v4f*)&xm[r * XM_STRIDE + k4] = o;
    }
  }
  __syncthreads();

  // ---- Phase 2: fp32 WMMA GEMM; wave wv owns 16x16 tile at cols wv*16 ----
  if (wv < NWAVES_MMA) {
    v8f acc = {};
    // A fragment: lanes 0-15 -> row=lane, K=k0/k0+1 ; lanes 16-31 -> K=k0+2/k0+3
    const float* ap = &xm[(lane & 15) * XM_STRIDE + 2 * (lane >> 4)];
    if (PREP) {
      const float2* bp = bfrag + ((size_t)wv * 96) * 32 + lane;
#pragma unroll 8
      for (int s = 0; s < KDIM / 4; ++s) {
        v2f a = *(const v2f*)(ap + s * 4);
        v2f b = *(const v2f*)(bp + (size_t)s * 32);
        acc = __builtin_amdgcn_wmma_f32_16x16x4_f32(
            false, a, false, b, (short)0, acc, false, false);
      }
    } else {
      const float* wp = mw + (size_t)(wv * 16 + (lane & 15)) * KDIM + 2 * (lane >> 4);
#pragma unroll 8
      for (int s = 0; s < KDIM / 4; ++s) {
        v2f a = *(const v2f*)(ap + s * 4);
        v2f b = *(const v2f*)(wp + s * 4);
        acc = __builtin_amdgcn_wmma_f32_16x16x4_f32(
            false, a, false, b, (short)0, acc, false, false);
      }
    }
    // C/D layout: VGPR v, lanes 0-15 -> M=v ; lanes 16-31 -> M=v+8
    int rbase = (lane >> 4) * 8;
    int col   = wv * 16 + (lane & 15);
#pragma unroll
    for (int v = 0; v < 8; ++v)
      xs[(rbase + v) * XS_STRIDE + col] = acc[v];
  }
  __syncthreads();

  // ---- Phase 3: xv = xs + ln(xs); xa = ln(xv); out = ln(xa). wave per row ----
  {
    const int r = wv;
    float v[6];
#pragma unroll
    for (int j = 0; j < 3; ++j) {
      v2f pv = *(const v2f*)&xs[r * XS_STRIDE + j * 64 + lane * 2];
      v[2 * j]     = pv[0];
      v[2 * j + 1] = pv[1];
    }
    ln_step192(v, lane, vssw, vssb, /*residual=*/true);
    ln_step192(v, lane, asw,  asb,  /*residual=*/false);
    ln_step192(v, lane, fnw,  fnb,  /*residual=*/false);
    float* op = out + (size_t)(row0 + r) * NDIM;
#pragma unroll
    for (int j = 0; j < 3; ++j) {
      v2f sv; sv[0] = v[2 * j]; sv[1] = v[2 * j + 1];
      __builtin_nontemporal_store(sv, (v2f*)(op + j * 64 + lane * 2));
    }
  }
}

extern "C" void kernel_launch(void* const* d_in, const int* in_sizes, int n_in,
                              void* d_out, int out_size, void* d_ws, size_t ws_size,
                              hipStream_t stream) {
  (void)in_sizes; (void)n_in; (void)out_size;
  const float* x    = (const float*)d_in[0];
  // d_in[1] = saliency_map : dead in the reference (result deleted) -> skipped
  const float* mw   = (const float*)d_in[2];
  const float* mnw  = (const float*)d_in[3];
  const float* mnb  = (const float*)d_in[4];
  const float* vssw = (const float*)d_in[5];
  const float* vssb = (const float*)d_in[6];
  const float* asw  = (const float*)d_in[7];
  const float* asb  = (const float*)d_in[8];
  const float* fnw  = (const float*)d_in[9];
  const float* fnb  = (const float*)d_in[10];
  float* out = (float*)d_out;

  const size_t bfrag_bytes = (size_t)NWAVES_MMA * 96 * 32 * sizeof(float2); // 294912
  const bool prep = (ws_size >= bfrag_bytes) && (d_ws != nullptr);

  dim3 grid(MROWS / MTILE);   // 6272 blocks
  if (prep) {
    prep_bfrag<<<NWAVES_MMA * 96, 32, 0, stream>>>(mw, (float2*)d_ws);
    fused_focus_mamba<true><<<grid, THREADS, 0, stream>>>(
        x, mw, (const float2*)d_ws, mnw, mnb, vssw, vssb, asw, asb, fnw, fnb, out);
  } else {
    fused_focus_mamba<false><<<grid, THREADS, 0, stream>>>(
        x, mw, nullptr, mnw, mnb, vssw, vssb, asw, asb, fnw, fnb, out);
  }
}